// GRNNTransformSimple_24438363914722
// MI455X (gfx1250) — compile-verified
//
#include <hip/hip_runtime.h>
#include <math.h>

typedef float v2f __attribute__((ext_vector_type(2)));
typedef float v8f __attribute__((ext_vector_type(8)));

#define NH   128        // hidden size
#define KDIM 384        // 3*NH
#define WHS  388        // padded LDS row stride for w_h (388 % 64 == 4 -> conflict-free)
#define US   132        // padded LDS row stride for u block
#define WAVES 8
#define ROWS_PER_WG (WAVES * 16)
#define LDS_FLOATS (NH*WHS + WAVES*16*US)   // 49664 + 16896 = 66560 floats = 266240 B

// ---------------- Leaf level (j = DEPTH): emb = tanh(c @ w_u^T + b_u) ----------------
__global__ void grnn_leaf(const float* __restrict__ cont,
                          const float* __restrict__ w_u,
                          const float* __restrict__ b_u,
                          float* __restrict__ out, int L)
{
  long idx = (long)blockIdx.x * blockDim.x + threadIdx.x;   // over L*128 outputs
  if (idx >= (long)L * NH) return;
  int  col = (int)(idx & (NH - 1));
  long row = idx >> 7;
  const float* cp = cont + row * 7;
  const float* wr = w_u + col * 7;
  float acc = b_u[col];
  #pragma unroll
  for (int i = 0; i < 7; ++i) acc = fmaf(wr[i], cp[i], acc);
  out[idx] = tanhf(acc);
}

// ---------------- Inner level: emb = tanh([h_L|h_R|u] @ w_h^T + b_h) ----------------
__global__ void __launch_bounds__(256, 1) grnn_level(
    const float* __restrict__ cont,      // contents for this level (pre-offset)
    const int*   __restrict__ child,     // children pairs for this level (pre-offset)
    const float* __restrict__ w_u,
    const float* __restrict__ b_u,
    const float* __restrict__ w_h,       // (128, 384) row-major
    const float* __restrict__ b_h,
    const float* __restrict__ emb_prev,  // (2L, 128)
    float*       __restrict__ emb_out,   // (L, 128)
    int Lrows,                            // L
    int mblocks)                          // ceil(L / 128)
{
  extern __shared__ float lds[];
  float* whs = lds;                 // 128 x WHS   (w_h, padded)
  float* us  = lds + NH * WHS;      // 8 waves x 16 x US (u block)

  const int tid  = threadIdx.x;     // 0..255
  const int wave = tid >> 5;
  const int lane = tid & 31;
  const int lrow = lane & 15;
  const int half = lane >> 4;

  // Stage w_h into LDS with padded stride (float4 copies, 16B aligned both sides).
  for (int i = tid; i < NH * (KDIM / 4); i += 256) {
    int r  = i / (KDIM / 4);
    int q4 = i - r * (KDIM / 4);
    float4 v = ((const float4*)w_h)[i];
    *(float4*)&whs[r * WHS + q4 * 4] = v;
  }
  __syncthreads();

  float* uw = us + wave * (16 * US);

  for (int mb = blockIdx.x; mb < mblocks; mb += gridDim.x) {
    const int m0 = mb * ROWS_PER_WG + wave * 16;   // this wave's 16 rows
    if (m0 >= Lrows) continue;                     // wave-uniform guard: EXEC stays all-1s
    const int myrow = m0 + lrow;                   // lanes l and l+16 share a row
    const int cl = child[2 * myrow + 0];
    const int cr = child[2 * myrow + 1];
    const float* hL = emb_prev + (long)cl * NH;
    const float* hR = emb_prev + (long)cr * NH;

    // ---- u = tanh(w_u @ c + b_u): lane computes row lrow, cols [half*64, half*64+64)
    float f0, f1, f2, f3, f4, f5, f6;
    {
      const float* cp = cont + (long)myrow * 7;
      f0 = cp[0]; f1 = cp[1]; f2 = cp[2]; f3 = cp[3]; f4 = cp[4]; f5 = cp[5]; f6 = cp[6];
    }
    for (int cc = 0; cc < 64; ++cc) {
      const int col = half * 64 + cc;
      const float* wr = w_u + col * 7;
      float a = b_u[col];
      a = fmaf(wr[0], f0, a); a = fmaf(wr[1], f1, a); a = fmaf(wr[2], f2, a);
      a = fmaf(wr[3], f3, a); a = fmaf(wr[4], f4, a); a = fmaf(wr[5], f5, a);
      a = fmaf(wr[6], f6, a);
      uw[lrow * US + col] = tanhf(a);
    }

    // ---- accumulators initialized with bias (C/D layout: lane -> N=t*16+lrow, VGPR r -> M=r+8*half)
    v8f acc[8];
    #pragma unroll
    for (int t = 0; t < 8; ++t) {
      float bv = b_h[t * 16 + lrow];
      #pragma unroll
      for (int r = 0; r < 8; ++r) acc[t][r] = bv;
    }

    // ---- K block 0: h_L (X columns 0..127), A fragment straight from global
    #pragma unroll 4
    for (int ks = 0; ks < 32; ++ks) {
      const int k0 = 4 * ks;
      v2f a = *(const v2f*)(hL + (k0 + 2 * half));
      #pragma unroll
      for (int t = 0; t < 8; ++t) {
        v2f b = *(const v2f*)&whs[(t * 16 + lrow) * WHS + (k0 + 2 * half)];
        acc[t] = __builtin_amdgcn_wmma_f32_16x16x4_f32(false, a, false, b,
                                                       (short)0, acc[t], false, false);
      }
    }
    // ---- K block 1: h_R (X columns 128..255)
    #pragma unroll 4
    for (int ks = 0; ks < 32; ++ks) {
      const int k0 = 4 * ks;
      v2f a = *(const v2f*)(hR + (k0 + 2 * half));
      #pragma unroll
      for (int t = 0; t < 8; ++t) {
        v2f b = *(const v2f*)&whs[(t * 16 + lrow) * WHS + (128 + k0 + 2 * half)];
        acc[t] = __builtin_amdgcn_wmma_f32_16x16x4_f32(false, a, false, b,
                                                       (short)0, acc[t], false, false);
      }
    }
    // ---- K block 2: u (X columns 256..383), A fragment from LDS (same-wave, DS in-order)
    #pragma unroll 4
    for (int ks = 0; ks < 32; ++ks) {
      const int k0 = 4 * ks;
      v2f a = *(const v2f*)&uw[lrow * US + (k0 + 2 * half)];
      #pragma unroll
      for (int t = 0; t < 8; ++t) {
        v2f b = *(const v2f*)&whs[(t * 16 + lrow) * WHS + (256 + k0 + 2 * half)];
        acc[t] = __builtin_amdgcn_wmma_f32_16x16x4_f32(false, a, false, b,
                                                       (short)0, acc[t], false, false);
      }
    }

    // ---- emb = tanh(acc), store
    #pragma unroll
    for (int t = 0; t < 8; ++t) {
      #pragma unroll
      for (int r = 0; r < 8; ++r) {
        const int row = m0 + r + 8 * half;
        emb_out[(long)row * NH + t * 16 + lrow] = tanhf(acc[t][r]);
      }
    }
  }
}

// ---------------- Host side ----------------
extern "C" void kernel_launch(void* const* d_in, const int* in_sizes, int n_in,
                              void* d_out, int out_size, void* d_ws, size_t ws_size,
                              hipStream_t stream)
{
  const float* contents = (const float*)d_in[0];
  const int*   children = (const int*)  d_in[1];
  const float* w_u      = (const float*)d_in[2];
  const float* b_u      = (const float*)d_in[3];
  const float* w_h      = (const float*)d_in[4];
  const float* b_h      = (const float*)d_in[5];

  // Ping-pong embedding buffers: even levels in bufs[0] (max 262144*128 f32),
  // odd levels in bufs[1] (max 131072*128 f32). Total ~201 MB of d_ws.
  float* bufs[2];
  bufs[0] = (float*)d_ws;
  bufs[1] = bufs[0] + (size_t)262144 * 128;

  const size_t shmem = (size_t)LDS_FLOATS * sizeof(float);   // 266240 B (< 320 KB/WGP)
  (void)hipFuncSetAttribute((const void*)grnn_level,
                            hipFuncAttributeMaxDynamicSharedMemorySize, (int)shmem);

  // Leaf level j = 12
  {
    int L = 64 << 12;
    size_t coff = (size_t)64 * ((1u << 12) - 1);             // OFFSETS[12] = 262080
    long total = (long)L * 128;
    int blocks = (int)((total + 255) / 256);
    grnn_leaf<<<blocks, 256, 0, stream>>>(contents + coff * 7, w_u, b_u, bufs[0], L);
  }

  // Inner levels j = 11 .. 0
  for (int j = 11; j >= 0; --j) {
    int L = 64 << j;
    size_t off = (size_t)64 * ((1u << j) - 1);               // OFFSETS[j] == INNER_OFF[j]
    int mblocks = (L + ROWS_PER_WG - 1) / ROWS_PER_WG;
    int nwg = mblocks < 2048 ? mblocks : 2048;
    const float* prev = bufs[(j + 1) & 1];
    float* outp = (j == 0) ? (float*)d_out : bufs[j & 1];
    grnn_level<<<nwg, 256, shmem, stream>>>(contents + off * 7, children + off * 2,
                                            w_u, b_u, w_h, b_h, prev, outp, L, mblocks);
  }
}